// MedRNN_24086176596075
// MI455X (gfx1250) — compile-verified
//
#include <hip/hip_runtime.h>
#include <hip/hip_bf16.h>
#include <math.h>
#include <stdint.h>

// Problem constants (from reference): B=128, T=64, V=1024, TS=64, H=512
#define BT   8192          // B*T flattened steps
#define KV   1024          // visit_emb inner dim
#define G3   1536          // 3*H gate rows
#define LDI  1088          // W_ih row stride (V+TS)
#define HD   512           // hidden dim

#define KC     64          // K-chunk staged per TDM transfer
#define NCHUNK (KV / KC)   // 16 chunks
#define LROW   68          // LDS row stride in dwords (64 data + 4 pad, conflict-free)

typedef __attribute__((ext_vector_type(2))) float v2f;
typedef __attribute__((ext_vector_type(8))) float v8f;
typedef __attribute__((ext_vector_type(4))) unsigned int u32x4;
typedef __attribute__((ext_vector_type(8))) int i32x8;
typedef __attribute__((ext_vector_type(4))) int i32x4;

// ---------------------------------------------------------------------------
// TDM: DMA one 16-row x 64-col f32 tile (row stride 1024 elements) from
// global memory into LDS at lds_byte_off, padding 4 dwords after each
// 64-dword row (LDS row stride = 68 dwords).  D# layout per CDNA5 ISA §8.
// ---------------------------------------------------------------------------
__device__ __forceinline__ void tdm_load_tile_16x64(const float* gsrc,
                                                    unsigned lds_byte_off) {
  unsigned long long ga = (unsigned long long)(uintptr_t)gsrc;
  u32x4 g0;
  g0[0] = 1u;                                   // count=1 (valid user D#)
  g0[1] = lds_byte_off;                         // lds_addr (bytes)
  g0[2] = (unsigned)ga;                         // global_addr[31:0]
  g0[3] = (unsigned)((ga >> 32) & 0x1FFFFFFu)   // global_addr[56:32]
          | 0x80000000u;                        // type=2 ("image")
  i32x8 g1;
  g1[0] = (int)0x07520000;   // data_size=4B | pad_en | pad_interval=64dw | pad=4dw
  g1[1] = (int)(1024u << 16); // tensor_dim0 = 1024 (cols)
  g1[2] = (int)(8192u << 16); // tensor_dim1 = 8192 (rows)
  g1[3] = (int)(64u << 16);   // tile_dim0 = 64
  g1[4] = (int)16;            // tile_dim1 = 16, tile_dim2 = 0
  g1[5] = (int)1024;          // tensor_dim0_stride = 1024
  g1[6] = 0;
  g1[7] = 0;
  i32x4 gz = {0, 0, 0, 0};
#if __has_include(<hip/amd_detail/amd_gfx1250_TDM.h>)
  i32x8 gz8 = {0, 0, 0, 0, 0, 0, 0, 0};
  __builtin_amdgcn_tensor_load_to_lds(g0, g1, gz, gz, gz8, 0);
#else
  __builtin_amdgcn_tensor_load_to_lds(g0, g1, gz, gz, 0);
#endif
}

// ---------------------------------------------------------------------------
// Kernel 0: fold time-embedding + bias into per-gate-row coefficients.
//   c1[g] = sum_ts W_ih[g, V+ts] * W_time[ts]
//   c0[g] = sum_ts W_ih[g, V+ts] * b_time[ts] + b_ih[g]
// ---------------------------------------------------------------------------
__global__ __launch_bounds__(256) void medrnn_coeffs(
    const float* __restrict__ W_time, const float* __restrict__ b_time,
    const float* __restrict__ W_ih,   const float* __restrict__ b_ih,
    float* __restrict__ c1, float* __restrict__ c0) {
  int g = blockIdx.x * blockDim.x + threadIdx.x;
  if (g >= G3) return;
  const float* wrow = W_ih + (size_t)g * LDI + KV;
  float s1 = 0.f, s0 = 0.f;
#pragma unroll 8
  for (int t = 0; t < 64; ++t) {
    float w = wrow[t];
    s1 += w * W_time[t];
    s0 += w * b_time[t];
  }
  c1[g] = s1;
  c0[g] = s0 + b_ih[g];
}

// ---------------------------------------------------------------------------
// Kernel 1: xg[m,g] = sum_k ve[m,k]*W_ih[g,k] + intervals[m]*c1[g] + c0[g]
// via V_WMMA_F32_16X16X4_F32.
//
// Block = 256 threads = 8 waves, covering one 16-row M-tile x 128 N-cols.
// The A tile (16 x KC of visit_emb) is shared by all 8 waves: wave 0 DMAs it
// into double-buffered LDS with the Tensor Data Mover while everyone computes
// on the previous chunk; waves read A fragments with ds_load_b64 (padded rows
// -> conflict-free broadcast).  B fragments are per-wave contiguous b64 loads.
//
// A frag (16x4 f32): lane l: M=(l&15), K = k + 2*(l>>4) + {0,1}
// B frag (4x16 f32): lane l: N=(l&15), K = k + 2*(l>>4) + {0,1} (B[k][n]=W_ih[n][k])
// C/D (16x16 f32):   vgpr v, lane l: M = v + 8*(l>>4), N = (l&15)
// ---------------------------------------------------------------------------
__global__ __launch_bounds__(256) void medrnn_xg_wmma(
    const float* __restrict__ ve,        // [BT, KV]
    const float* __restrict__ W_ih,      // [G3, LDI]
    const float* __restrict__ intervals, // [BT]
    const float* __restrict__ c1, const float* __restrict__ c0,
    float* __restrict__ xg) {            // [BT, G3]
  __shared__ float lA[2][16 * LROW];     // double-buffered padded A tile

  const int wave = threadIdx.x >> 5;     // 0..7
  const int lane = threadIdx.x & 31;
  const int bm   = blockIdx.x / (G3 / 128);   // 512 M-tiles
  const int bn   = blockIdx.x % (G3 / 128);   // 12 N-groups
  const int tm   = bm * 16;
  const int tn   = bn * 128 + wave * 16;

  const int r  = lane & 15;
  const int kh = lane >> 4;              // 0 or 1

  const float* bPtr = W_ih + (size_t)(tn + r) * LDI + 2 * kh;
  const float* aRow = ve + (size_t)tm * KV;    // block's A-tile origin

  // Prologue: DMA chunk 0 into buffer 0.
  if (wave == 0) {
    tdm_load_tile_16x64(aRow, (unsigned)(uintptr_t)&lA[0][0]);
    __builtin_amdgcn_s_wait_tensorcnt(0);
  }
  __syncthreads();

  v8f acc = {};
  for (int c = 0; c < NCHUNK; ++c) {
    const int cur = c & 1;
    // Kick off next chunk's DMA into the other buffer (freed by the barrier
    // at the end of the previous iteration).
    if (wave == 0 && (c + 1) < NCHUNK)
      tdm_load_tile_16x64(aRow + (c + 1) * KC,
                          (unsigned)(uintptr_t)&lA[cur ^ 1][0]);

    const float* Ab = &lA[cur][r * LROW + 2 * kh];
    const float* Bb = bPtr + c * KC;
#pragma unroll
    for (int kc = 0; kc < KC; kc += 4) {
      v2f a = *(const v2f*)(Ab + kc);          // ds_load_b64 (broadcast tile)
      v2f b = *(const v2f*)(Bb + kc);          // global_load_b64
      acc = __builtin_amdgcn_wmma_f32_16x16x4_f32(
          /*neg_a=*/false, a, /*neg_b=*/false, b,
          /*c_mod=*/(short)0, acc, /*reuse_a=*/false, /*reuse_b=*/false);
    }
    __syncthreads();                     // all waves done reading lA[cur]
    if (wave == 0)
      __builtin_amdgcn_s_wait_tensorcnt(0);    // chunk c+1 landed in LDS
    __syncthreads();                     // release everyone onto lA[cur^1]
  }

  // Epilogue: fold rank-1 time-embedding term + bias, store 16x16 tile.
  const int n = tn + r;
  const float c1n = c1[n];
  const float c0n = c0[n];
#pragma unroll
  for (int v = 0; v < 8; ++v) {
    int m = tm + v + 8 * kh;
    xg[(size_t)m * G3 + n] = acc[v] + intervals[m] * c1n + c0n;
  }
}

// ---------------------------------------------------------------------------
// Kernel 2: sequential GRU scan over all 8192 steps.
// Single persistent workgroup (512 threads = 16 waves on one WGP).
// Hidden state h[512] lives in LDS; thread g owns gate rows g, H+g, 2H+g.
// W_hh (3 MB) stays L2-resident across all 8192 matvec passes.
// ---------------------------------------------------------------------------
__global__ __launch_bounds__(512) void medrnn_scan(
    const float* __restrict__ xg,    // [BT, G3]
    const float* __restrict__ W_hh,  // [G3, HD]
    const float* __restrict__ b_hh,  // [G3]
    const int*   __restrict__ lens,  // [B]
    float* __restrict__ out) {       // [B, HD]
  __shared__ float hs[HD];
  const int g = threadIdx.x;
  hs[g] = 0.0f;

  const float br = b_hh[g];
  const float bz = b_hh[HD + g];
  const float bn = b_hh[2 * HD + g];
  const float4* wr = (const float4*)(W_hh + (size_t)g * HD);
  const float4* wz = (const float4*)(W_hh + (size_t)(HD + g) * HD);
  const float4* wn = (const float4*)(W_hh + (size_t)(2 * HD + g) * HD);
  __syncthreads();

  for (int s = 0; s < BT; ++s) {
    const float* xrow = xg + (size_t)s * G3;
    // prefetch next step's gate inputs while we do this step's matvec
    __builtin_prefetch(xrow + G3 + g, 0, 1);

    float ar = br, az = bz, an = bn;
    const float4* hv = (const float4*)hs;
#pragma unroll 4
    for (int j = 0; j < HD / 4; ++j) {
      float4 h4 = hv[j];
      float4 r4 = wr[j];
      ar += h4.x * r4.x + h4.y * r4.y + h4.z * r4.z + h4.w * r4.w;
      float4 z4 = wz[j];
      az += h4.x * z4.x + h4.y * z4.y + h4.z * z4.z + h4.w * z4.w;
      float4 n4 = wn[j];
      an += h4.x * n4.x + h4.y * n4.y + h4.z * n4.z + h4.w * n4.w;
    }

    float rg = 1.0f / (1.0f + __expf(-(xrow[g] + ar)));
    float zg = 1.0f / (1.0f + __expf(-(xrow[HD + g] + az)));
    float ng = tanhf(xrow[2 * HD + g] + rg * an);
    float hp = hs[g];
    float hnew = (1.0f - zg) * ng + zg * hp;
    if ((s & 63) >= lens[s >> 6]) hnew = hp;   // masked (padded) timestep

    __syncthreads();          // everyone done reading old h
    hs[g] = hnew;
    __syncthreads();          // new h visible before next step

    if ((s & 63) == 63) out[(size_t)(s >> 6) * HD + g] = hnew;
  }
}

// ---------------------------------------------------------------------------
extern "C" void kernel_launch(void* const* d_in, const int* in_sizes, int n_in,
                              void* d_out, int out_size, void* d_ws, size_t ws_size,
                              hipStream_t stream) {
  const float* ve        = (const float*)d_in[0]; // visit_emb [128,64,1024]
  const float* intervals = (const float*)d_in[1]; // [128,64]
  const float* W_time    = (const float*)d_in[2]; // [64,1]
  const float* b_time    = (const float*)d_in[3]; // [64]
  const float* W_ih      = (const float*)d_in[4]; // [1536,1088]
  const float* W_hh      = (const float*)d_in[5]; // [1536,512]
  const float* b_ih      = (const float*)d_in[6]; // [1536]
  const float* b_hh      = (const float*)d_in[7]; // [1536]
  const int*   lens      = (const int*)d_in[8];   // [128]
  float* out = (float*)d_out;                     // [128,512]

  float* xg = (float*)d_ws;                       // 8192*1536 floats = 48 MB
  float* c1 = xg + (size_t)BT * G3;               // 1536 floats
  float* c0 = c1 + G3;                            // 1536 floats

  medrnn_coeffs<<<(G3 + 255) / 256, 256, 0, stream>>>(W_time, b_time, W_ih, b_ih, c1, c0);

  // 512 M-tiles x 12 N-groups = 6144 blocks, 8 waves (16x16 tiles) per block
  medrnn_xg_wmma<<<6144, 256, 0, stream>>>(ve, W_ih, intervals, c1, c0, xg);

  medrnn_scan<<<1, 512, 0, stream>>>(xg, W_hh, b_hh, lens, out);
}